// GCNMagnetModel_68393059221808
// MI455X (gfx1250) — compile-verified
//
#include <hip/hip_runtime.h>
#include <hip/hip_bf16.h>
#include <math.h>

typedef __attribute__((ext_vector_type(2))) float v2f;
typedef __attribute__((ext_vector_type(8))) float v8f;

#define HDIM 128
#define NGRAPHS 64

// ---------------- degree / normalization ----------------
__global__ void k_init_deg(float* deg, int N) {
  int i = blockIdx.x * blockDim.x + threadIdx.x;
  if (i < N) deg[i] = 1.0f;  // self loop
}

__global__ void k_edge_deg(const int* __restrict__ dst, float* deg, int E) {
  int e = blockIdx.x * blockDim.x + threadIdx.x;
  if (e < E) atomicAdd(&deg[dst[e]], 1.0f);
}

__global__ void k_rsqrt(float* deg, int N) {
  int i = blockIdx.x * blockDim.x + threadIdx.x;
  if (i < N) deg[i] = rsqrtf(deg[i]);
}

// ---------------- layer 1 transform (K=2, scalar) ----------------
__global__ void k_l1(const float* __restrict__ x, const float* __restrict__ W1,
                     float* __restrict__ hw, int NH) {
  int i = blockIdx.x * blockDim.x + threadIdx.x;
  if (i >= NH) return;
  int n = i >> 7;
  int j = i & (HDIM - 1);
  hw[i] = x[2 * n] * W1[j] + x[2 * n + 1] * W1[HDIM + j];
}

// ---------------- WMMA f32 GEMM: out[N,128] = A[N,128] @ W[128,128] ----------------
__global__ __launch_bounds__(256) void k_gemm_wmma(const float* __restrict__ A,
                                                   const float* __restrict__ W,
                                                   float* __restrict__ out,
                                                   int mtiles) {
  __shared__ float sW[HDIM * HDIM];  // 64 KB
  {
    const float4* Wv = (const float4*)W;
    float4* sWv = (float4*)sW;
    for (int i = threadIdx.x; i < (HDIM * HDIM) / 4; i += 256) sWv[i] = Wv[i];
  }
  __syncthreads();

  int wave = threadIdx.x >> 5;
  int lane = threadIdx.x & 31;
  int mtile = blockIdx.x * 8 + wave;
  if (mtile >= mtiles) return;  // wave-uniform: EXEC stays all-ones for WMMA

  int m0 = mtile << 4;
  int r = lane & 15;            // M index (A) / N index (B,C,D) within tile
  int kh = (lane >> 4) << 1;    // K sub-offset: lanes 0-15 -> K{0,1}, lanes 16-31 -> K{2,3}

  const float* arow = A + (size_t)(m0 + r) * HDIM;

  v8f acc[8];
  v8f zero = {0.f, 0.f, 0.f, 0.f, 0.f, 0.f, 0.f, 0.f};
#pragma unroll
  for (int t = 0; t < 8; ++t) acc[t] = zero;

  for (int k = 0; k < HDIM; k += 4) {
    v2f a;
    a.x = arow[k + kh];
    a.y = arow[k + kh + 1];
#pragma unroll
    for (int t = 0; t < 8; ++t) {
      int col = (t << 4) + r;
      v2f b;
      b.x = sW[(k + kh) * HDIM + col];
      b.y = sW[(k + kh + 1) * HDIM + col];
      acc[t] = __builtin_amdgcn_wmma_f32_16x16x4_f32(false, a, false, b,
                                                     (short)0, acc[t], false, false);
    }
  }

  // C/D layout: lanes 0-15 hold rows 0..7 (vgpr v -> row v), lanes 16-31 rows 8..15
  int rb = (lane >> 4) << 3;
#pragma unroll
  for (int t = 0; t < 8; ++t) {
#pragma unroll
    for (int v = 0; v < 8; ++v) {
      out[(size_t)(m0 + rb + v) * HDIM + (t << 4) + r] = acc[t][v];
    }
  }
}

// ---------------- agg init: hw*dinv^2 + b ----------------
__global__ void k_init_agg(const float* __restrict__ hw, const float* __restrict__ dinv,
                           const float* __restrict__ b, float* __restrict__ agg, int NH) {
  int i = blockIdx.x * blockDim.x + threadIdx.x;
  if (i >= NH) return;
  int n = i >> 7;
  int j = i & (HDIM - 1);
  float dv = dinv[n];
  agg[i] = hw[i] * dv * dv + b[j];
}

// ---------------- edge scatter: one wave per edge ----------------
__global__ void k_scatter(const int* __restrict__ src, const int* __restrict__ dst,
                          const float* __restrict__ dinv, const float* __restrict__ hw,
                          float* agg, int E) {
  int e = blockIdx.x * 8 + (threadIdx.x >> 5);
  if (e >= E) return;
  int lane = threadIdx.x & 31;
  int s = src[e];
  int d = dst[e];
  float nm = dinv[s] * dinv[d];
  const float4* hp = (const float4*)(hw + (size_t)s * HDIM);
  float4 v = hp[lane];
  float* ap = agg + (size_t)d * HDIM + (lane << 2);
  atomicAdd(ap + 0, v.x * nm);
  atomicAdd(ap + 1, v.y * nm);
  atomicAdd(ap + 2, v.z * nm);
  atomicAdd(ap + 3, v.w * nm);
}

// ---------------- activation ----------------
__global__ void k_tanh(const float* __restrict__ agg, float* __restrict__ h, int NH) {
  int i = blockIdx.x * blockDim.x + threadIdx.x;
  if (i < NH) h[i] = tanhf(agg[i]);
}

// ---------------- pooling ----------------
__global__ void k_zero_cnt(int* cnt) {
  int i = threadIdx.x;
  if (i < NGRAPHS) cnt[i] = 0;
}

__global__ void k_count(const int* __restrict__ batch, int* cnt, int N) {
  int i = blockIdx.x * blockDim.x + threadIdx.x;
  if (i < N) atomicAdd(&cnt[batch[i]], 1);
}

__global__ void k_offsets(const int* __restrict__ cnt, int* start) {
  if (threadIdx.x == 0 && blockIdx.x == 0) {
    int s = 0;
    start[0] = 0;
    for (int g = 0; g < NGRAPHS; ++g) { s += cnt[g]; start[g + 1] = s; }
  }
}

__global__ void k_pool(const float* __restrict__ h, const int* __restrict__ start,
                       const int* __restrict__ cnt, float* __restrict__ pooled) {
  int g = blockIdx.x;       // 64 blocks
  int j = threadIdx.x;      // 128 threads
  int s = start[g], e = start[g + 1];
  int c = cnt[g];
  float mx = -3.0e38f;
  float sum = 0.0f;
  for (int n = s; n < e; ++n) {
    float v = h[(size_t)n * HDIM + j];
    mx = fmaxf(mx, v);
    sum += v;
  }
  pooled[g * (2 * HDIM) + j] = (c > 0) ? mx : 0.0f;
  pooled[g * (2 * HDIM) + HDIM + j] = sum / fmaxf((float)c, 1.0f);
}

// ---------------- output head: [64,256] @ [256,41] + bo, tanh ----------------
__global__ void k_out(const float* __restrict__ pooled, const float* __restrict__ Wo,
                      const float* __restrict__ bo, float* __restrict__ out) {
  int g = blockIdx.x;   // 64 blocks
  int o = threadIdx.x;  // 64 threads, guard to 41
  if (o >= 41) return;
  float acc = bo[o];
  const float* p = pooled + g * (2 * HDIM);
  for (int r = 0; r < 2 * HDIM; ++r) acc += p[r] * Wo[r * 41 + o];
  out[g * 41 + o] = tanhf(acc);
}

extern "C" void kernel_launch(void* const* d_in, const int* in_sizes, int n_in,
                              void* d_out, int out_size, void* d_ws, size_t ws_size,
                              hipStream_t stream) {
  const float* x  = (const float*)d_in[0];
  const int* ei   = (const int*)d_in[1];
  const int* batch= (const int*)d_in[2];
  const float* W1 = (const float*)d_in[3];
  const float* b1 = (const float*)d_in[4];
  const float* W2 = (const float*)d_in[5];
  const float* b2 = (const float*)d_in[6];
  const float* W3 = (const float*)d_in[7];
  const float* b3 = (const float*)d_in[8];
  const float* Wo = (const float*)d_in[9];
  const float* bo = (const float*)d_in[10];

  const int N = in_sizes[0] / 2;
  const int E = in_sizes[1] / 2;
  const int NH = N * HDIM;
  const int mtiles = (N + 15) / 16;

  const int* srcp = ei;
  const int* dstp = ei + E;

  // workspace carve-out
  char* ws = (char*)d_ws;
  size_t off = 0;
  auto carve = [&](size_t bytes) -> void* {
    void* p = ws + off;
    off = (off + bytes + 255) & ~(size_t)255;
    return p;
  };
  float* dinv   = (float*)carve((size_t)N * 4);
  float* hw     = (float*)carve((size_t)NH * 4);
  float* agg    = (float*)carve((size_t)NH * 4);
  float* h      = (float*)carve((size_t)NH * 4);
  int*   cnt    = (int*)carve(NGRAPHS * 4);
  int*   startb = (int*)carve((NGRAPHS + 1) * 4);
  float* pooled = (float*)carve(NGRAPHS * 2 * HDIM * 4);
  (void)ws_size; (void)n_in; (void)out_size;

  dim3 b256(256);
  int gN  = (N + 255) / 256;
  int gE  = (E + 255) / 256;
  int gNH = (NH + 255) / 256;
  int gEw = (E + 7) / 8;          // one wave per edge, 8 waves/block
  int gMt = (mtiles + 7) / 8;     // 8 wave-tiles per block

  // degree + rsqrt
  k_init_deg<<<gN, b256, 0, stream>>>(dinv, N);
  k_edge_deg<<<gE, b256, 0, stream>>>(dstp, dinv, E);
  k_rsqrt<<<gN, b256, 0, stream>>>(dinv, N);

  // layer 1
  k_l1<<<gNH, b256, 0, stream>>>(x, W1, hw, NH);
  k_init_agg<<<gNH, b256, 0, stream>>>(hw, dinv, b1, agg, NH);
  k_scatter<<<gEw, b256, 0, stream>>>(srcp, dstp, dinv, hw, agg, E);
  k_tanh<<<gNH, b256, 0, stream>>>(agg, h, NH);

  // layer 2
  k_gemm_wmma<<<gMt, b256, 0, stream>>>(h, W2, hw, mtiles);
  k_init_agg<<<gNH, b256, 0, stream>>>(hw, dinv, b2, agg, NH);
  k_scatter<<<gEw, b256, 0, stream>>>(srcp, dstp, dinv, hw, agg, E);
  k_tanh<<<gNH, b256, 0, stream>>>(agg, h, NH);

  // layer 3
  k_gemm_wmma<<<gMt, b256, 0, stream>>>(h, W3, hw, mtiles);
  k_init_agg<<<gNH, b256, 0, stream>>>(hw, dinv, b3, agg, NH);
  k_scatter<<<gEw, b256, 0, stream>>>(srcp, dstp, dinv, hw, agg, E);
  k_tanh<<<gNH, b256, 0, stream>>>(agg, h, NH);

  // pooling + head
  k_zero_cnt<<<1, 64, 0, stream>>>(cnt);
  k_count<<<gN, b256, 0, stream>>>(batch, cnt, N);
  k_offsets<<<1, 1, 0, stream>>>(cnt, startb);
  k_pool<<<NGRAPHS, HDIM, 0, stream>>>(h, startb, cnt, pooled);
  k_out<<<NGRAPHS, 64, 0, stream>>>(pooled, Wo, bo, (float*)d_out);
}